// StreamingDurationProjector_51616916963919
// MI455X (gfx1250) — compile-verified
//
#include <hip/hip_runtime.h>
#include <stdint.h>

// Problem constants (match the JAX reference).
#define B_DIM 2048
#define T_DIM 2048
#define ROWS 32          // rows per block == wave32 width; lane = row
#define CT 32            // timesteps per chunk
#define CTP 36           // padded row: 144B -> 16B aligned for b128, 2-way bank conflict max
#define NCHUNK (T_DIM / CT)
#define NTHREADS 256     // 8 waves: all stage data, wave 0 scans

#if __has_builtin(__builtin_amdgcn_global_load_async_to_lds_b32)
#define HAVE_ASYNC_LDS 1
#else
#define HAVE_ASYNC_LDS 0
#endif

typedef int v4i __attribute__((vector_size(16)));
typedef __attribute__((address_space(1))) int glb_i32;
typedef __attribute__((address_space(3))) int lds_i32;
typedef __attribute__((address_space(1))) v4i glb_v4i;
typedef __attribute__((address_space(3))) v4i lds_v4i;

__device__ __forceinline__ void async_ld_b32(const float* g, float* l) {
#if HAVE_ASYNC_LDS
  __builtin_amdgcn_global_load_async_to_lds_b32(
      (glb_i32*)const_cast<float*>(g), (lds_i32*)l, 0, 0);
#else
  *l = *g;
#endif
}

// 16B global -> 16B LDS per lane; 512B per wave per instruction, as 4 rows x 128B.
__device__ __forceinline__ void async_ld_b128(const float* g, float* l) {
#if HAVE_ASYNC_LDS && __has_builtin(__builtin_amdgcn_global_load_async_to_lds_b128)
  __builtin_amdgcn_global_load_async_to_lds_b128(
      (glb_v4i*)const_cast<float*>(g), (lds_v4i*)l, 0, 0);
#else
#pragma unroll
  for (int j = 0; j < 4; ++j) async_ld_b32(g + j, l + j);
#endif
}

__device__ __forceinline__ void wait_async_lds() {
#if HAVE_ASYNC_LDS
#if __has_builtin(__builtin_amdgcn_s_wait_asynccnt)
  __builtin_amdgcn_s_wait_asynccnt(0);
#else
  asm volatile("s_wait_asynccnt 0" ::: "memory");
#endif
#endif
}

__global__ __launch_bounds__(NTHREADS)
void StreamingDurationProjector_kernel(
    const float* __restrict__ row_exec,
    const float* __restrict__ row_rnd,
    const float* __restrict__ row_bud,
    const float* __restrict__ row_sp,
    const float* __restrict__ row_co,
    const float* __restrict__ row_bd,
    const float* __restrict__ row_pf,
    const float* __restrict__ carry_init,
    const float* __restrict__ off_init,
    float* __restrict__ out)
{
  // Double-buffered input tiles, single-buffered output tile. 78,336 B total.
  __shared__ __align__(16) float s_in[2][7][ROWS][CTP];
  __shared__ __align__(16) float s_out[3][ROWS][CTP];

  const int tid  = threadIdx.x;
  const int row0 = blockIdx.x * ROWS;
  const bool scanner = (tid < ROWS);

  const float* in[7] = {row_exec, row_rnd, row_bud, row_sp, row_co, row_bd, row_pf};

  float* proj = out;
  float* bhit = out + (size_t)B_DIM * T_DIM;
  float* bdec = out + 2 * (size_t)B_DIM * T_DIM;

  // Per-thread 16B segment within the [ROWS x CT] tile: 32 rows x 8 segs = 256.
  const int seg_r = tid >> 3;        // row 0..31
  const int seg_c = (tid & 7) * 4;   // column start 0,4,...,28

  // Cooperative, coalesced async stage of one tile per input into LDS.
  auto load_chunk = [&](int k, int bi) {
    const int t0 = k * CT;
#pragma unroll
    for (int a = 0; a < 7; ++a) {
      const float* src = in[a] + (size_t)row0 * T_DIM + t0;
      async_ld_b128(src + (size_t)seg_r * T_DIM + seg_c, &s_in[bi][a][seg_r][seg_c]);
    }
  };

  // Per-lane recurrent state (lives in wave-0 registers for the whole row).
  float carry = 0.0f, off = 0.0f;
  if (scanner) {
    carry = carry_init[row0 + tid];
    off   = off_init[row0 + tid];
  }

  load_chunk(0, 0);
  wait_async_lds();
  __syncthreads();

  for (int k = 0; k < NCHUNK; ++k) {
    const int bi = k & 1;
    if (k + 1 < NCHUNK) load_chunk(k + 1, bi ^ 1);  // stream next chunk behind the scan

    if (scanner) {
      const int l = tid;  // lane == row within tile
#pragma unroll 4
      for (int c = 0; c < CT; ++c) {
        const float vex = s_in[bi][0][l][c];
        const float vrn = s_in[bi][1][l][c];
        const float vbu = s_in[bi][2][l][c];
        const float vsp = s_in[bi][3][l][c];
        const float vco = s_in[bi][4][l][c];
        const float vbd = s_in[bi][5][l][c];
        const float vpf = s_in[bi][6][l][c];

        const bool active = (vsp > 0.5f) | (vco > 0.5f);
        const float anchor  = fmaxf(vrn, 1.0f);
        const float banchor = fmaxf(vbu, 1.0f);
        const float total   = fmaxf(vex + carry, 0.0f);
        float frames = fmaxf(floorf(total + 0.5f), 1.0f);
        const float lower = fmaxf(ceilf(banchor - (3.0f + off)), 1.0f);
        const float upper = fmaxf(floorf(banchor + (3.0f - off)), lower);
        frames = fminf(fmaxf(frames, lower), upper);
        float new_off   = off + (frames - banchor);
        float new_carry = total - frames;
        const bool bevent = (vbd >= 0.5f) | (vpf > 0.5f);
        // CARRY_DECAY = OFFSET_DECAY = 0.5 (< 1), so the decay branch is live.
        const float dec_carry = new_carry * 0.5f;
        const float dec_off   = fminf(fmaxf(new_off * 0.5f, -3.0f), 3.0f);
        new_carry = bevent ? dec_carry : new_carry;
        new_off   = bevent ? dec_off   : new_off;
        const float hit = (active & bevent) ? 1.0f : 0.0f;  // bhit == bdecay here

        s_out[0][l][c] = active ? frames : anchor;
        s_out[1][l][c] = hit;
        s_out[2][l][c] = hit;

        carry = active ? new_carry : carry;
        off   = active ? new_off   : off;
      }
    }
    __syncthreads();  // s_out ready; s_in[bi] fully consumed

    // Coalesced 16B store-back of the output tile by all 8 waves.
    {
      const int t0 = k * CT;
#pragma unroll
      for (int o = 0; o < 3; ++o) {
        float* dst = (o == 0 ? proj : (o == 1 ? bhit : bdec)) + (size_t)row0 * T_DIM + t0;
        const float4 v = *(const float4*)&s_out[o][seg_r][seg_c];
        *(float4*)(dst + (size_t)seg_r * T_DIM + seg_c) = v;
      }
    }

    wait_async_lds();   // chunk k+1 landed in s_in[bi^1]
    __syncthreads();    // make it visible to wave 0
  }

  if (scanner) {
    out[3 * (size_t)B_DIM * T_DIM + row0 + tid] = carry;           // final carry
    out[3 * (size_t)B_DIM * T_DIM + B_DIM + row0 + tid] = off;     // final offset
  }
}

extern "C" void kernel_launch(void* const* d_in, const int* in_sizes, int n_in,
                              void* d_out, int out_size, void* d_ws, size_t ws_size,
                              hipStream_t stream) {
  (void)in_sizes; (void)n_in; (void)d_ws; (void)ws_size; (void)out_size;
  const float* row_exec = (const float*)d_in[0];
  const float* row_rnd  = (const float*)d_in[1];
  const float* row_bud  = (const float*)d_in[2];
  const float* row_sp   = (const float*)d_in[3];
  const float* row_co   = (const float*)d_in[4];
  const float* row_bd   = (const float*)d_in[5];
  const float* row_pf   = (const float*)d_in[6];
  const float* ci       = (const float*)d_in[7];
  const float* oi       = (const float*)d_in[8];
  float* out = (float*)d_out;

  dim3 grid(B_DIM / ROWS);   // 64 blocks x 256 threads
  dim3 block(NTHREADS);
  StreamingDurationProjector_kernel<<<grid, block, 0, stream>>>(
      row_exec, row_rnd, row_bud, row_sp, row_co, row_bd, row_pf, ci, oi, out);
}